// PointClassifier_52140902974018
// MI455X (gfx1250) — compile-verified
//
#include <hip/hip_runtime.h>
#include <math.h>

typedef float v2f __attribute__((ext_vector_type(2)));
typedef float v8f __attribute__((ext_vector_type(8)));

#define NPTS   4096
#define NBATCH 8
#define KNN    10
#define R2     0.04f
#define FEATD  40

// ---------------------------------------------------------------------------
// Kernel 1: ball query + feature build.
// One block = 256 query points; full batch point cloud staged in LDS (48KB).
// feat[b][i][4k..4k+3] = [d2, x, y, z] of k-th (ascending index) neighbor.
// ---------------------------------------------------------------------------
__global__ void ball_query_feat_kernel(const float* __restrict__ points,
                                       float* __restrict__ feat) {
  __shared__ float px[NPTS], py[NPTS], pz[NPTS];
  const int b      = blockIdx.x >> 4;   // 16 blocks per batch
  const int blkInB = blockIdx.x & 15;
  const float* P = points + (size_t)b * NPTS * 3;
  for (int j = threadIdx.x; j < NPTS; j += 256) {
    px[j] = P[j * 3 + 0];
    py[j] = P[j * 3 + 1];
    pz[j] = P[j * 3 + 2];
  }
  __syncthreads();

  const int i = blkInB * 256 + threadIdx.x;
  const float xi = px[i], yi = py[i], zi = pz[i];
  float* g = feat + ((size_t)b * NPTS + i) * FEATD;
#pragma unroll
  for (int c = 0; c < FEATD; ++c) g[c] = 0.0f;   // missing neighbors -> 0

  int count = 0;
  for (int j = 0; j < NPTS; ++j) {
    const float dx = px[j] - xi, dy = py[j] - yi, dz = pz[j] - zi;
    const float d2 = dx * dx + dy * dy + dz * dz;
    if (d2 < R2) {
      g[4 * count + 0] = d2;
      g[4 * count + 1] = px[j];
      g[4 * count + 2] = py[j];
      g[4 * count + 3] = pz[j];
      if (++count == KNN) break;
    }
  }
}

// ---------------------------------------------------------------------------
// WMMA fp32 dual-M-tile layer: out(32 x dout) = relu(in(32 x din) * W^T + b)
// using V_WMMA_F32_16X16X4_F32. W is (dout, din) row-major. Each B operand
// (weights) is loaded once and reused for two WMMAs (rows 0-15 and 16-31),
// halving weight traffic and doubling WMMA density per vmem op.
// Must be called by a full wave (EXEC all ones).
// ---------------------------------------------------------------------------
__device__ __forceinline__ void wmma_layer2(const float* in, int sIn, int din,
                                            const float* __restrict__ W,
                                            const float* __restrict__ bias,
                                            int dout, float* out, int sOut,
                                            bool relu, int l16, int half) {
  const int ntiles = (dout + 15) >> 4;
  for (int nt = 0; nt < ntiles; ++nt) {
    const int col = nt * 16 + l16;
    const bool colOK = col < dout;
    const float* wrow = W + (size_t)(colOK ? col : 0) * din;
    v8f acc0 = {}, acc1 = {};
    for (int k0 = 0; k0 < din; k0 += 4) {
      const int kk = k0 + half * 2;
      v2f bv;
      bv.x = colOK ? wrow[kk] : 0.0f;
      bv.y = colOK ? wrow[kk + 1] : 0.0f;
      v2f a0, a1;
      a0.x = in[l16 * sIn + kk];
      a0.y = in[l16 * sIn + kk + 1];
      a1.x = in[(16 + l16) * sIn + kk];
      a1.y = in[(16 + l16) * sIn + kk + 1];
      acc0 = __builtin_amdgcn_wmma_f32_16x16x4_f32(
          false, a0, false, bv, (short)0, acc0, false, false);
      acc1 = __builtin_amdgcn_wmma_f32_16x16x4_f32(
          false, a1, false, bv, (short)0, acc1, false, false);
    }
    const float bb = colOK ? bias[col] : 0.0f;
#pragma unroll
    for (int r = 0; r < 8; ++r) {
      const int row = r + half * 8;
      float v0 = acc0[r] + bb;
      float v1 = acc1[r] + bb;
      if (relu) { v0 = fmaxf(v0, 0.0f); v1 = fmaxf(v1, 0.0f); }
      if (colOK) {
        out[row * sOut + nt * 16 + l16]        = v0;
        out[(16 + row) * sOut + nt * 16 + l16] = v1;
      }
    }
  }
}

// ---------------------------------------------------------------------------
// Kernel 2: fused cluster MLP 40->256->128->64->32->8->1 (ReLU between).
// One wave per block owning 32 points (two 16-row WMMA tiles). The feature
// tile is staged global->LDS with GLOBAL_LOAD_ASYNC_TO_LDS_B128 (ASYNCcnt),
// then activations ping-pong through padded LDS buffers.
// ---------------------------------------------------------------------------
#define SA 130   // bufA stride (>=128, padded vs 64 banks)
#define SB 258   // bufB stride (>=256, padded)
#define ROWS 32  // rows (points) per wave/block

__global__ void cluster_mlp_kernel(const float* __restrict__ feat,
                                   const float* __restrict__ W1, const float* __restrict__ b1,
                                   const float* __restrict__ W2, const float* __restrict__ b2,
                                   const float* __restrict__ W3, const float* __restrict__ b3,
                                   const float* __restrict__ W4, const float* __restrict__ b4,
                                   const float* __restrict__ W5, const float* __restrict__ b5,
                                   const float* __restrict__ W6, const float* __restrict__ b6,
                                   float* __restrict__ pw) {
  // bufF: 32x40 contiguous (async-copy target), bufA: 32x130, bufB: 32x258
  __shared__ float smem[ROWS * FEATD + ROWS * SA + ROWS * SB];   // ~54.8 KB
  const int lane = threadIdx.x & 31;
  const int l16  = lane & 15;
  const int half = lane >> 4;
  const int rowBase = blockIdx.x * ROWS;

  float* bufF = smem;
  float* bufA = bufF + ROWS * FEATD;
  float* bufB = bufA + ROWS * SA;

  // ---- async stage of the 32x40 feature tile (1280 floats, contiguous) ----
  {
    const float* gsrc = feat + (size_t)rowBase * FEATD;
    const unsigned ldsBase = (unsigned)(uintptr_t)bufF;  // low 32b = LDS offset
#pragma unroll
    for (int t = 0; t < (ROWS * FEATD) / (32 * 4); ++t) {  // 10 iterations
      const int e = t * 32 + lane;                         // 16-byte chunk id
      const unsigned dst = ldsBase + e * 16;
      const float* src = gsrc + e * 4;
      asm volatile("global_load_async_to_lds_b128 %0, %1, off"
                   :: "v"(dst), "v"(src)
                   : "memory");
    }
    asm volatile("s_wait_asynccnt 0" ::: "memory");
  }
  __syncthreads();

  wmma_layer2(bufF, FEATD, 40, W1, b1, 256, bufB, SB, true, l16, half);
  __syncthreads();
  wmma_layer2(bufB, SB, 256,  W2, b2, 128, bufA, SA, true, l16, half);
  __syncthreads();
  wmma_layer2(bufA, SA, 128,  W3, b3, 64,  bufB, SB, true, l16, half);
  __syncthreads();
  wmma_layer2(bufB, SB, 64,   W4, b4, 32,  bufA, SA, true, l16, half);
  __syncthreads();
  wmma_layer2(bufA, SA, 32,   W5, b5, 8,   bufB, SB, true, l16, half);
  __syncthreads();

  // final 8 -> 1 (scalar per point, one lane per row)
  {
    float v = b6[0];
#pragma unroll
    for (int k = 0; k < 8; ++k) v += bufB[lane * SB + k] * W6[k];
    pw[rowBase + lane] = v;
  }
}

// ---------------------------------------------------------------------------
// Kernel 3: apply permutation  pwp[b][i] = pw[b][perm[i]]
// ---------------------------------------------------------------------------
__global__ void permute_kernel(const float* __restrict__ pw,
                               const int* __restrict__ perm,
                               float* __restrict__ pwp) {
  const int idx = blockIdx.x * blockDim.x + threadIdx.x;   // 0..32767
  const int b = idx >> 12, i = idx & (NPTS - 1);
  pwp[idx] = pw[(b << 12) + perm[i]];
}

// ---------------------------------------------------------------------------
// Kernel 4: one clf MLP layer. M=8 batch rows padded to a 16-row WMMA tile.
// One wave per block; block n handles output columns [16n, 16n+16).
// ---------------------------------------------------------------------------
__global__ void clf_layer_kernel(const float* __restrict__ in,
                                 const float* __restrict__ W,
                                 const float* __restrict__ bias,
                                 float* __restrict__ out,
                                 int din, int dout, int relu) {
  const int lane = threadIdx.x & 31;
  const int l16  = lane & 15;
  const int half = lane >> 4;
  const int col  = blockIdx.x * 16 + l16;
  const bool colOK = col < dout;
  const bool rowOK = l16 < NBATCH;
  const float* wrow = W + (size_t)(colOK ? col : 0) * din;

  v8f acc = {};
  for (int k0 = 0; k0 < din; k0 += 4) {
    const int kk = k0 + half * 2;
    v2f a, bv;
    a.x = rowOK ? in[l16 * din + kk] : 0.0f;
    a.y = rowOK ? in[l16 * din + kk + 1] : 0.0f;
    bv.x = colOK ? wrow[kk] : 0.0f;
    bv.y = colOK ? wrow[kk + 1] : 0.0f;
    acc = __builtin_amdgcn_wmma_f32_16x16x4_f32(
        false, a, false, bv, (short)0, acc, false, false);
  }
  const float bb = colOK ? bias[col] : 0.0f;
#pragma unroll
  for (int r = 0; r < 8; ++r) {
    const int row = r + half * 8;
    if (row < NBATCH && colOK) {
      float v = acc[r] + bb;
      if (relu) v = fmaxf(v, 0.0f);
      out[row * dout + col] = v;
    }
  }
}

// ---------------------------------------------------------------------------
// Kernel 5: softmax over 10 classes, one thread per batch row.
// ---------------------------------------------------------------------------
__global__ void softmax_kernel(const float* __restrict__ logits,
                               float* __restrict__ out) {
  const int t = threadIdx.x;
  if (t < NBATCH) {
    float m = -1e30f;
#pragma unroll
    for (int c = 0; c < 10; ++c) m = fmaxf(m, logits[t * 10 + c]);
    float e[10], s = 0.0f;
#pragma unroll
    for (int c = 0; c < 10; ++c) { e[c] = __expf(logits[t * 10 + c] - m); s += e[c]; }
    const float inv = 1.0f / s;
#pragma unroll
    for (int c = 0; c < 10; ++c) out[t * 10 + c] = e[c] * inv;
  }
}

// ---------------------------------------------------------------------------
extern "C" void kernel_launch(void* const* d_in, const int* in_sizes, int n_in,
                              void* d_out, int out_size, void* d_ws, size_t ws_size,
                              hipStream_t stream) {
  const float* points = (const float*)d_in[0];
  const int*   perm   = (const int*)d_in[1];
  // cluster_params: 6 x (W, b)
  const float* W1 = (const float*)d_in[2];   const float* b1 = (const float*)d_in[3];
  const float* W2 = (const float*)d_in[4];   const float* b2 = (const float*)d_in[5];
  const float* W3 = (const float*)d_in[6];   const float* b3 = (const float*)d_in[7];
  const float* W4 = (const float*)d_in[8];   const float* b4 = (const float*)d_in[9];
  const float* W5 = (const float*)d_in[10];  const float* b5 = (const float*)d_in[11];
  const float* W6 = (const float*)d_in[12];  const float* b6 = (const float*)d_in[13];
  // clf_params: 5 x (W, b)
  const float* CW0 = (const float*)d_in[14]; const float* cb0 = (const float*)d_in[15];
  const float* CW1 = (const float*)d_in[16]; const float* cb1 = (const float*)d_in[17];
  const float* CW2 = (const float*)d_in[18]; const float* cb2 = (const float*)d_in[19];
  const float* CW3 = (const float*)d_in[20]; const float* cb3 = (const float*)d_in[21];
  const float* CW4 = (const float*)d_in[22]; const float* cb4 = (const float*)d_in[23];

  const int M = NBATCH * NPTS;               // 32768 points total
  float* feat   = (float*)d_ws;              // M * 40
  float* pw     = feat + (size_t)M * FEATD;  // M
  float* pwp    = pw + M;                    // M
  float* act1   = pwp + M;                   // 8 * 2048
  float* act2   = act1 + NBATCH * 2048;      // 8 * 1024
  float* act3   = act2 + NBATCH * 1024;      // 8 * 512
  float* act4   = act3 + NBATCH * 512;       // 8 * 256
  float* logits = act4 + NBATCH * 256;       // 8 * 10

  ball_query_feat_kernel<<<NBATCH * (NPTS / 256), 256, 0, stream>>>(points, feat);
  cluster_mlp_kernel<<<M / ROWS, 32, 0, stream>>>(feat,
      W1, b1, W2, b2, W3, b3, W4, b4, W5, b5, W6, b6, pw);
  permute_kernel<<<M / 256, 256, 0, stream>>>(pw, perm, pwp);

  clf_layer_kernel<<<2048 / 16, 32, 0, stream>>>(pwp,  CW0, cb0, act1, 4096, 2048, 1);
  clf_layer_kernel<<<1024 / 16, 32, 0, stream>>>(act1, CW1, cb1, act2, 2048, 1024, 1);
  clf_layer_kernel<<< 512 / 16, 32, 0, stream>>>(act2, CW2, cb2, act3, 1024,  512, 1);
  clf_layer_kernel<<< 256 / 16, 32, 0, stream>>>(act3, CW3, cb3, act4,  512,  256, 1);
  clf_layer_kernel<<<        1, 32, 0, stream>>>(act4, CW4, cb4, logits, 256,  10, 0);

  softmax_kernel<<<1, 32, 0, stream>>>(logits, (float*)d_out);
}